// JpegLayer_75359496175875
// MI455X (gfx1250) — compile-verified
//
#include <hip/hip_runtime.h>

typedef float v2f __attribute__((ext_vector_type(2)));
typedef float v8f __attribute__((ext_vector_type(8)));

#define WAVES_PER_BLOCK 8
#define SM_STRIDE 17   // 16 + 1 padding -> conflict-free LDS transpose bounce

// rgb:  [32,3,512,512] f32
// dct:  [8,8] f32 (D, applied as D @ X @ D^T)
// out:  [32,3,64,64,8,8] f32
__global__ __launch_bounds__(256) void jpeg_dct_wmma(
    const float* __restrict__ rgb,
    const float* __restrict__ dct,
    float* __restrict__ out)
{
  __shared__ float sm[WAVES_PER_BLOCK][16 * SM_STRIDE];

  const int tid  = threadIdx.x;
  const int wave = tid >> 5;
  const int lane = tid & 31;
  const int c    = lane & 15;   // column index inside 16x16 tile
  const int h    = lane >> 4;   // lane half (ISA f32 fragment layout)

  // one wave == one 16x16 tile == four 8x8 DCT blocks
  const int t    = blockIdx.x * WAVES_PER_BLOCK + wave;   // 0 .. 32767
  const int b    = t >> 10;       // batch (1024 tiles per 512x512 image)
  const int tile = t & 1023;
  const int ty   = tile >> 5;     // tile row 0..31
  const int tx   = tile & 31;     // tile col 0..31

  // ---- constant fragments of block-diagonal D2 = diag(D, D).
  // A-chunk layout (pass 1)  : lane holds D2[c][4*k0 + 2h + v]
  // B-chunk layout of D2^T (pass 2): lane holds D2^T[4*k0+2h+v][c] = D2[c][...]
  // => identical fragments serve both passes.
  v2f dfrag[4];
  #pragma unroll
  for (int k0 = 0; k0 < 4; ++k0) {
    #pragma unroll
    for (int v = 0; v < 2; ++v) {
      const int j = 4 * k0 + 2 * h + v;
      dfrag[k0][v] = ((c >> 3) == (j >> 3)) ? dct[(c & 7) * 8 + (j & 7)] : 0.0f;
    }
  }

  // ---- load 16x16 tile of all 3 channels directly in B-fragment layout:
  // xf[ch][k0][v] = rgb[b][ch][row0 + 4*k0 + 2h + v][col0 + c]
  const int row0 = ty * 16;
  const int col0 = tx * 16;
  float xf[3][4][2];
  #pragma unroll
  for (int ch = 0; ch < 3; ++ch) {
    const float* src =
        rgb + (((size_t)(b * 3 + ch) * 512 + row0) * 512 + col0 + c);
    #pragma unroll
    for (int k0 = 0; k0 < 4; ++k0) {
      #pragma unroll
      for (int v = 0; v < 2; ++v) {
        const int row = 4 * k0 + 2 * h + v;
        xf[ch][k0][v] = src[(size_t)row * 512];
      }
    }
  }

  // ---- RGB -> YCbCr, offset, clamp [0,1], center by -128/255
  const float xm[3][3] = {
      { 0.299f,         0.587f,        0.114f       },
      { -0.168735892f, -0.331264108f,  0.5f         },
      { 0.5f,          -0.418687589f, -0.081312411f } };
  const float off[3] = { 0.0f, 128.0f / 255.0f, 128.0f / 255.0f };
  const float ctr = 128.0f / 255.0f;

  #pragma unroll
  for (int ch = 0; ch < 3; ++ch) {
    // B fragments for this output channel
    v2f bfrag[4];
    #pragma unroll
    for (int k0 = 0; k0 < 4; ++k0) {
      #pragma unroll
      for (int v = 0; v < 2; ++v) {
        float y = xm[ch][0] * xf[0][k0][v]
                + xm[ch][1] * xf[1][k0][v]
                + xm[ch][2] * xf[2][k0][v] + off[ch];
        y = fminf(fmaxf(y, 0.0f), 1.0f) - ctr;
        bfrag[k0][v] = y;
      }
    }

    // ---- pass 1: P = D2 * X   (four K=4 chunks accumulated)
    v8f p = {};
    #pragma unroll
    for (int k0 = 0; k0 < 4; ++k0) {
      p = __builtin_amdgcn_wmma_f32_16x16x4_f32(
          /*neg_a=*/false, dfrag[k0],
          /*neg_b=*/false, bfrag[k0],
          /*c_mod=*/(short)0, p,
          /*reuse_a=*/false, /*reuse_b=*/false);
    }

    // ---- bounce P (C layout) through LDS to build pass-2 A fragments
    #pragma unroll
    for (int r = 0; r < 8; ++r) {
      sm[wave][(r + 8 * h) * SM_STRIDE + c] = p[r];
    }
    asm volatile("s_wait_dscnt 0x0" ::: "memory");

    v2f afrag[4];
    #pragma unroll
    for (int k0 = 0; k0 < 4; ++k0) {
      #pragma unroll
      for (int v = 0; v < 2; ++v) {
        afrag[k0][v] = sm[wave][c * SM_STRIDE + 4 * k0 + 2 * h + v];
      }
    }

    // ---- pass 2: Q = P * D2^T  (B fragment of D2^T == dfrag)
    v8f q = {};
    #pragma unroll
    for (int k0 = 0; k0 < 4; ++k0) {
      q = __builtin_amdgcn_wmma_f32_16x16x4_f32(
          false, afrag[k0],
          false, dfrag[k0],
          (short)0, q, false, false);
    }

    // ---- store: Q[M][N] -> out[b][ch][2*ty + M/8][2*tx + N/8][M%8][N%8]
    // per lane: M = r + 8h, N = c  =>  bh = 2*ty+h, bw = 2*tx + (c>>3)
    const int bh = 2 * ty + h;
    const int bw = 2 * tx + (c >> 3);
    float* dst = out + ((((size_t)(b * 3 + ch) * 64 + bh) * 64 + bw) * 64
                        + (c & 7));
    #pragma unroll
    for (int r = 0; r < 8; ++r) {
      dst[r * 8] = q[r];
    }
  }
}

extern "C" void kernel_launch(void* const* d_in, const int* in_sizes, int n_in,
                              void* d_out, int out_size, void* d_ws, size_t ws_size,
                              hipStream_t stream) {
  const float* rgb = (const float*)d_in[0];   // [32,3,512,512]
  const float* dct = (const float*)d_in[1];   // [8,8]
  float* out = (float*)d_out;                 // [32,3,64,64,8,8]

  // 32 images * (32*32) tiles = 32768 waves; 8 waves (256 threads) per block
  dim3 grid(4096), block(256);
  hipLaunchKernelGGL(jpeg_dct_wmma, grid, block, 0, stream, rgb, dct, out);
}